// RoIAlign_73693048865536
// MI455X (gfx1250) — compile-verified
//
#include <hip/hip_runtime.h>
#include <stdint.h>

// ---- problem constants (match reference) ----
#define RATIO_   (1.0f/32.0f)
#define FH       50
#define FW       50
#define NC       256
#define HOUT     7
#define WOUT     7
#define NS       2
// LDS patch capacity: roi width <= 7.25 feat px -> <=10 cols; per-ph y span <= 3 rows
#define XS_MAX   10
#define YS_MAX   4

// ---- CDNA5 async global->LDS path (guarded; falls back to plain copy) ----
#if __has_builtin(__builtin_amdgcn_global_load_async_to_lds_b32)
#define HAVE_ASYNC_LDS 1
#else
#define HAVE_ASYNC_LDS 0
#endif

#if __has_builtin(__builtin_amdgcn_global_load_async_to_lds_b128)
#define HAVE_ASYNC_LDS128 1
#else
#define HAVE_ASYNC_LDS128 0
#endif

#if __has_builtin(__builtin_amdgcn_s_wait_asynccnt)
#define WAIT_ASYNC() __builtin_amdgcn_s_wait_asynccnt(0)
#else
#define WAIT_ASYNC() asm volatile("s_wait_asynccnt 0x0" ::: "memory")
#endif

typedef int v4i __attribute__((ext_vector_type(4)));

typedef __attribute__((address_space(1))) int GI;   // global int (probe: b32 takes int*)
typedef __attribute__((address_space(3))) int LI;   // LDS int
typedef __attribute__((address_space(1))) v4i GV4;  // global int4 (probe: b128 takes v4i*)
typedef __attribute__((address_space(3))) v4i LV4;  // LDS int4

__device__ __forceinline__ void async_copy_f32(const float* g, float* l) {
#if HAVE_ASYNC_LDS
  int* gg = (int*)g;   // drop const + retype in generic AS
  int* ll = (int*)l;
  __builtin_amdgcn_global_load_async_to_lds_b32((GI*)gg, (LI*)ll, 0, 0);
#else
  *l = *g;   // plain vmem load + ds store fallback (same-wave RAW is in-order on DS)
#endif
}

__device__ __forceinline__ void async_copy_b128(const float* g, float* l) {
#if HAVE_ASYNC_LDS128
  v4i* gg = (v4i*)g;
  v4i* ll = (v4i*)l;
  __builtin_amdgcn_global_load_async_to_lds_b128((GV4*)gg, (LV4*)ll, 0, 0);
#else
  float4 v = *(const float4*)g;
  *(float4*)l = v;
#endif
}

__device__ __forceinline__ size_t faddr(int transposed, int b, int y, int x, int c) {
  return transposed ? ((((size_t)b * FH + y) * FW + x) * NC + c)
                    : ((((size_t)b * NC + c) * FH + y) * FW + x);
}

// ---------------- transpose (B,C,H,W) -> (B,H,W,C), tiled through LDS ----------------
__global__ void transpose_bchw_to_bhwc(const float* __restrict__ in, float* __restrict__ out) {
  __shared__ float tile[32][FW + 1];          // 32 channels x 50 w, padded
  const int blk = blockIdx.x;                 // ((b*FH + h)*8 + ct)
  const int ct  = blk & 7;                    // NC/32 = 8 channel tiles
  const int bh  = blk >> 3;
  const int h   = bh % FH;
  const int b   = bh / FH;
  const int c0  = ct * 32;
  const int tid = threadIdx.x;

  // coalesced-in-w loads, staged asynchronously into LDS
  for (int idx = tid; idx < 32 * FW; idx += 256) {
    int cl = idx / FW, w = idx % FW;
    const float* g = in + ((((size_t)b * NC + (c0 + cl)) * FH + h) * FW + w);
    async_copy_f32(g, &tile[cl][w]);
  }
  WAIT_ASYNC();
  __syncthreads();                            // cross-wave sharing in the tile

  // coalesced-in-c stores
  for (int idx = tid; idx < FW * 32; idx += 256) {
    int w = idx / 32, cl = idx % 32;
    out[(((size_t)b * FH + h) * FW + w) * NC + c0 + cl] = tile[cl][w];
  }
}

// ---------------- main RoIAlign kernel: one block per (n, ph), lane = channel ----------------
__global__ void roialign_kernel(const float* __restrict__ feats,
                                const float* __restrict__ rois,
                                float* __restrict__ out,
                                int transposed) {
  __shared__ float patch[YS_MAX * XS_MAX * NC];   // 40 KB
  const int blk = blockIdx.x;
  const int ph  = blk % HOUT;
  const int n   = blk / HOUT;
  const int c   = threadIdx.x;

  const float* roi = rois + (size_t)n * 5;
  __builtin_prefetch(roi, 0, 0);                  // global_prefetch_b8

  const int b = (int)roi[0];
  float bx1 = fminf(fmaxf(roi[1] * RATIO_, 0.f), (float)FW);
  float by1 = fminf(fmaxf(roi[2] * RATIO_, 0.f), (float)FH);
  float bx2 = fminf(fmaxf(roi[3] * RATIO_, 0.f), (float)FW);
  float by2 = fminf(fmaxf(roi[4] * RATIO_, 0.f), (float)FH);
  const bool roi_valid = (bx2 - bx1 > 0.f) && (by2 - by1 > 0.f);
  const float bin_w = (bx2 - bx1) / (float)WOUT;
  const float bin_h = (by2 - by1) / (float)HOUT;

  // ---- y samples for this ph (shared across all 7 bins of the row) ----
  const float y1u = by1 + (float)ph * bin_h;
  const float y2b = fminf(fmaxf(y1u + bin_h, 0.f), (float)FH);
  const float y1b = fminf(fmaxf(y1u, 0.f), (float)FH);
  const bool yvalid = (y2b > y1b);

  int   yl[NS], yh[NS];
  float wyb[NS], wyt[NS];
  int ylo = FH - 1, yhi = 0;
  for (int s = 0; s < NS; ++s) {
    float py = y1b + ((float)s + 0.5f) * (bin_h / (float)NS);
    int l  = (int)floorf(py);
    l      = min(max(l, 0), FH - 1);
    int hh = min(l + 1, FH - 1);
    yl[s] = l; yh[s] = hh;
    wyb[s] = py - (float)l;     // reference: wyb = py - yl
    wyt[s] = (float)hh - py;    // reference: wyt = yh - py
    ylo = min(ylo, l); yhi = max(yhi, hh);
  }

  // ---- x bin bases + patch x-bounds ----
  float x1barr[WOUT];
  bool  xvalid[WOUT];
  int xlo = FW - 1, xhi = 0;
  for (int j = 0; j < WOUT; ++j) {
    float x1u = bx1 + (float)j * bin_w;
    float x2b = fminf(fmaxf(x1u + bin_w, 0.f), (float)FW);
    float x1b = fminf(fmaxf(x1u, 0.f), (float)FW);
    x1barr[j] = x1b;
    xvalid[j] = (x2b > x1b);
    for (int s = 0; s < NS; ++s) {
      float px = x1b + ((float)s + 0.5f) * (bin_w / (float)NS);
      int l  = min(max((int)floorf(px), 0), FW - 1);
      int hh = min(l + 1, FW - 1);
      xlo = min(xlo, l); xhi = max(xhi, hh);
    }
  }

  const int spanx = xhi - xlo + 1;
  const int spany = yhi - ylo + 1;
  const bool use_lds = (spanx >= 1) && (spany >= 1) && (spanx <= XS_MAX) && (spany <= YS_MAX);

  // ---- stage the (spany x spanx x 256ch) patch into LDS with async loads ----
  if (use_lds) {
    const int npos = spany * spanx;                 // <= 40 positions, 1 KB each
    if (transposed) {
      // b128 staging: 64 lanes x 16B cover one (y,x) position; 4 thread-groups
      // of the block cover 4 positions per iteration -> ~10 async ops per wave.
      const int grp = c >> 6;                       // 0..3: which position in the quad
      const int c4  = (c & 63) << 2;                // channel quad start 0,4,...,252
      for (int p0 = 0; p0 < npos; p0 += 4) {
        int p = p0 + grp;
        if (p < npos) {
          int dy = p / spanx, dx = p - dy * spanx;
          const float* g = feats +
              ((((size_t)b * FH + (ylo + dy)) * FW + (xlo + dx)) * NC + c4);
          async_copy_b128(g, &patch[p * NC + c4]);
        }
      }
      WAIT_ASYNC();        // each wave drains its own async loads...
      __syncthreads();     // ...then all waves' LDS writes are visible block-wide
    } else {
      // channels not contiguous in memory: b32, lane stages its own channel only
      int p = 0;
      for (int dy = 0; dy < spany; ++dy) {
        for (int dx = 0; dx < spanx; ++dx, ++p) {
          const float* g = feats + faddr(0, b, ylo + dy, xlo + dx, c);
          async_copy_f32(g, &patch[p * NC + c]);
        }
      }
      WAIT_ASYNC();        // lane re-reads only its own channel -> no barrier needed
    }
  }

  // ---- compute 7 bins, max over 2x2 samples, mask, store ----
  const size_t obase = (((size_t)n * NC + c) * HOUT + ph) * WOUT;
  for (int j = 0; j < WOUT; ++j) {
    float m = -__builtin_inff();
    for (int sx = 0; sx < NS; ++sx) {
      float px = x1barr[j] + ((float)sx + 0.5f) * (bin_w / (float)NS);
      int xl  = min(max((int)floorf(px), 0), FW - 1);
      int xh  = min(xl + 1, FW - 1);
      float wxh = px - (float)xl;   // reference names
      float wxl = (float)xh - px;
      for (int sy = 0; sy < NS; ++sy) {
        float vtl, vtr, vbl, vbr;
        if (use_lds) {
          int ry = yl[sy] - ylo, rb = yh[sy] - ylo;
          int rl = xl - xlo,     rr = xh - xlo;
          vtl = patch[(ry * spanx + rl) * NC + c];
          vtr = patch[(ry * spanx + rr) * NC + c];
          vbl = patch[(rb * spanx + rl) * NC + c];
          vbr = patch[(rb * spanx + rr) * NC + c];
        } else {
          vtl = feats[faddr(transposed, b, yl[sy], xl, c)];
          vtr = feats[faddr(transposed, b, yl[sy], xh, c)];
          vbl = feats[faddr(transposed, b, yh[sy], xl, c)];
          vbr = feats[faddr(transposed, b, yh[sy], xh, c)];
        }
        float top = wxh * vtr + wxl * vtl;   // match reference op order
        float bot = wxh * vbr + wxl * vbl;
        float val = wyb[sy] * bot + wyt[sy] * top;
        m = fmaxf(m, val);
      }
    }
    bool mask = roi_valid && yvalid && xvalid[j];
    out[obase + j] = mask ? m : 0.0f;
  }
}

extern "C" void kernel_launch(void* const* d_in, const int* in_sizes, int n_in,
                              void* d_out, int out_size, void* d_ws, size_t ws_size,
                              hipStream_t stream) {
  const float* features = (const float*)d_in[0];
  const float* rois     = (const float*)d_in[1];
  float*       out      = (float*)d_out;

  const int N = in_sizes[1] / 5;                          // 1024 RoIs
  const int B = in_sizes[0] / (NC * FH * FW);             // 4

  const size_t needed = (size_t)B * FH * FW * NC * sizeof(float);  // 10.24 MB
  const int transposed = (d_ws != nullptr && ws_size >= needed) ? 1 : 0;

  const float* feats = features;
  if (transposed) {
    float* t = (float*)d_ws;
    transpose_bchw_to_bhwc<<<B * FH * (NC / 32), 256, 0, stream>>>(features, t);
    feats = t;
  }
  roialign_kernel<<<N * HOUT, 256, 0, stream>>>(feats, rois, out, transposed);
}